// PillarQueryAndGroup_17884243821423
// MI455X (gfx1250) — compile-verified
//
#include <hip/hip_runtime.h>

// PillarQueryAndGroup for MI455X (gfx1250).
// Pure gather/concat workload -> bandwidth bound; use CDNA5 async
// global<->LDS data movers for the dominant 256B row copies.
//
// Output layout (flat float32, reference return order):
//   [0, 3M)                      pillar_indices (int -> float)
//   [3M, 3M+L)                   pillar_set_indices (int -> float)
//   [3M+L, 3M+L+L*(C+6))         group_features
//   [3M+L+L*(C+6)]               indice2bev = 1.0f

#define PCR_X0 (0.0f)
#define PCR_Y0 (-40.0f)
#define PCR_Z0 (-3.0f)

#define PQG_UNROLL 8   // rows in flight per wave between asynccnt drains

// ---------------------------------------------------------------------------
// Kernel 1: async feature-row gather-copy (C == 64 fast path).
// One wave32 moves one 64-float row per async op: 32 lanes x 8B = 256B.
// global -> LDS -> global entirely on the async engine (ASYNCcnt), no VGPR
// data movement. 8 rows (2KB) in flight per wave; 8 waves/block -> 16KB LDS.
// ---------------------------------------------------------------------------
__global__ void __launch_bounds__(256)
pqg_feat_copy_async(const float* __restrict__ pf,      // (N,64)
                    const int*   __restrict__ pt_idx,  // (L,)
                    float*       __restrict__ out_feat,// (L,70)
                    long long L)
{
    const int lane      = threadIdx.x & 31;
    const int waveInBlk = threadIdx.x >> 5;
    const long long wavesPerBlk = blockDim.x >> 5;
    const long long nWaves = (long long)gridDim.x * wavesPerBlk;
    const long long gwave  = (long long)blockIdx.x * wavesPerBlk + waveInBlk;
    // LDS staging: PQG_UNROLL slots x 256B per wave, lane strides of 8B.
    const unsigned ldsBase =
        (unsigned)waveInBlk * (PQG_UNROLL * 256u) + (unsigned)lane * 8u;

    for (long long base = gwave * PQG_UNROLL; base < L;
         base += nWaves * PQG_UNROLL) {
        // Stage: PQG_UNROLL async gathers, row k -> LDS slot k.
#pragma unroll
        for (int k = 0; k < PQG_UNROLL; ++k) {
            long long r = base + k; if (r >= L) r = L - 1;   // clamp: benign dup
            const int p = pt_idx[r];
            const float* src = pf + (long long)p * 64 + lane * 2;
            const unsigned lo = ldsBase + (unsigned)k * 256u;
            asm volatile("global_load_async_to_lds_b64 %0, %1, off"
                         :: "v"(lo), "v"(src) : "memory");
        }
        // LDS writes must land before the async stores read them.
        asm volatile("s_wait_asynccnt 0x0" ::: "memory");
        // Drain: PQG_UNROLL async stores, LDS slot k -> out row r (stride 70).
        // NT hint: output is write-once; keep L2 for the feature table.
#pragma unroll
        for (int k = 0; k < PQG_UNROLL; ++k) {
            long long r = base + k; if (r >= L) r = L - 1;
            float* dst = out_feat + r * 70 + lane * 2;
            const unsigned lo = ldsBase + (unsigned)k * 256u;
            asm volatile("global_store_async_from_lds_b64 %0, %1, off th:TH_STORE_NT"
                         :: "v"(dst), "v"(lo) : "memory");
        }
        // Stores must finish reading LDS before slots are reused.
        asm volatile("s_wait_asynccnt 0x0" ::: "memory");
    }
}

// ---------------------------------------------------------------------------
// Kernel 1b: generic fallback for C != 64 (plain element gather).
// ---------------------------------------------------------------------------
__global__ void __launch_bounds__(256)
pqg_feat_copy_generic(const float* __restrict__ pf,
                      const int*   __restrict__ pt_idx,
                      float*       __restrict__ out_feat,
                      long long L, int C)
{
    long long t = (long long)blockIdx.x * blockDim.x + threadIdx.x;
    long long tot = L * (long long)C;
    if (t >= tot) return;
    long long r = t / C;
    int       c = (int)(t - r * C);
    const int p = pt_idx[r];
    __builtin_nontemporal_store(pf[(long long)p * C + c],
                                out_feat + r * (long long)(C + 6) + c);
}

// ---------------------------------------------------------------------------
// Kernel 2: per-pair tail — cols [C, C+6) = [xyz+offset, xyz-center],
// plus pillar_set_indices int->float passthrough.
// ---------------------------------------------------------------------------
__global__ void __launch_bounds__(256)
pqg_tail6(const float* __restrict__ xyz,      // (N,3)
          const float* __restrict__ centers,  // (M,3)
          const int*   __restrict__ pt_idx,   // (L,)
          const int*   __restrict__ pl_idx,   // (L,)
          float*       __restrict__ out_feat, // (L,C+6)
          float*       __restrict__ out_psi,  // (L,)
          long long L, int C)
{
    long long r = (long long)blockIdx.x * blockDim.x + threadIdx.x;
    if (r >= L) return;
    const int p = pt_idx[r];
    const int q = pl_idx[r];
    const float px = xyz[(long long)p * 3 + 0];
    const float py = xyz[(long long)p * 3 + 1];
    const float pz = xyz[(long long)p * 3 + 2];
    const float cx = centers[(long long)q * 3 + 0];
    const float cy = centers[(long long)q * 3 + 1];
    const float cz = centers[(long long)q * 3 + 2];

    float* o = out_feat + r * (long long)(C + 6) + C;
    __builtin_nontemporal_store(px + PCR_X0, o + 0);   // absolute coords
    __builtin_nontemporal_store(py + PCR_Y0, o + 1);
    __builtin_nontemporal_store(pz + PCR_Z0, o + 2);
    __builtin_nontemporal_store(px - cx,     o + 3);   // rel to pillar center
    __builtin_nontemporal_store(py - cy,     o + 4);
    __builtin_nontemporal_store(pz - cz,     o + 5);

    __builtin_nontemporal_store((float)q, out_psi + r);
}

// ---------------------------------------------------------------------------
// Kernel 3: pillar_indices int->float passthrough + indice2bev flag.
// ---------------------------------------------------------------------------
__global__ void __launch_bounds__(256)
pqg_passthru(const int* __restrict__ pil_idx, // (M*3,)
             float*     __restrict__ out_pi,  // (M*3,)
             float*     __restrict__ out_flag,// (1,)
             long long n)
{
    long long i = (long long)blockIdx.x * blockDim.x + threadIdx.x;
    if (i < n) __builtin_nontemporal_store((float)pil_idx[i], out_pi + i);
    if (i == 0) out_flag[0] = 1.0f;
}

// ---------------------------------------------------------------------------
extern "C" void kernel_launch(void* const* d_in, const int* in_sizes, int n_in,
                              void* d_out, int out_size, void* d_ws, size_t ws_size,
                              hipStream_t stream)
{
    const float* xyz     = (const float*)d_in[0];  // (N,3)
    const float* pf      = (const float*)d_in[1];  // (N,C)
    const float* pc      = (const float*)d_in[2];  // (M,3)
    const int*   pil_idx = (const int*)d_in[3];    // (M,3)
    const int*   pt_set  = (const int*)d_in[4];    // (L,)
    const int*   pl_set  = (const int*)d_in[5];    // (L,)

    const long long N  = (long long)in_sizes[0] / 3;
    const int       C  = (int)((long long)in_sizes[1] / N);
    const long long M3 = (long long)in_sizes[3];   // M*3
    const long long L  = (long long)in_sizes[4];

    float* out      = (float*)d_out;
    float* out_pi   = out;                                   // M*3
    float* out_psi  = out + M3;                              // L
    float* out_feat = out + M3 + L;                          // L*(C+6)
    float* out_flag = out + M3 + L + L * (long long)(C + 6); // 1

    // Passthrough pillar_indices + flag.
    {
        int blocks = (int)((M3 + 255) / 256);
        pqg_passthru<<<blocks, 256, 0, stream>>>(pil_idx, out_pi, out_flag, M3);
    }

    // Dominant traffic: 256B feature rows via async global<->LDS movers.
    if (C == 64) {
        const int blocks = 4096;                 // 32768 wave32s, grid-stride
        const size_t dynLds = 8 /*waves*/ * PQG_UNROLL * 256; // 16KB
        pqg_feat_copy_async<<<blocks, 256, dynLds, stream>>>(
            pf, pt_set, out_feat, L);
    } else {
        long long tot = L * (long long)C;
        int blocks = (int)((tot + 255) / 256);
        pqg_feat_copy_generic<<<blocks, 256, 0, stream>>>(pf, pt_set, out_feat, L, C);
    }

    // 6 computed floats per pair + pillar_set_indices cast.
    {
        int blocks = (int)((L + 255) / 256);
        pqg_tail6<<<blocks, 256, 0, stream>>>(xyz, pc, pt_set, pl_set,
                                              out_feat, out_psi, L, C);
    }
}